// RecurrentRGCN_45543833206863
// MI455X (gfx1250) — compile-verified
//
#include <hip/hip_runtime.h>
#include <hip/hip_bf16.h>

// ---------------------------------------------------------------------------
// RecurrentRGCN layer for MI455X (gfx1250, wave32).
//
// Factorization: (h[src]+rel[etype]) @ W == (h@W)[src] + (rel@W)[etype]
//   -> edge phase is gather+scale+atomic-scatter (L2-resident, bandwidth bound)
//   -> compute reduces to node-level GEMMs done with V_WMMA_F32_16X16X4_F32,
//      with the 16 weight columns each wave needs held in registers (32 x v2f)
//      so the K loop is one b64 A-load + WMMA per step.
// ---------------------------------------------------------------------------

typedef __attribute__((ext_vector_type(2))) float v2f;
typedef __attribute__((ext_vector_type(8))) float v8f;

#define D 128
#define KSTEPS 32              // 128 / 4
#define RRELU_SLOPE 0.2291666666666667f

// -------------------- K0: per-row inverse L2 norm ---------------------------
__global__ void k_invnorm(const float* __restrict__ ent,
                          float* __restrict__ invn, int nrows) {
  int row  = blockIdx.x * 8 + (threadIdx.x >> 5);
  int lane = threadIdx.x & 31;
  if (row >= nrows) return;
  const float4* p = (const float4*)(ent + (size_t)row * D);
  float4 v = p[lane];
  float ss = v.x * v.x + v.y * v.y + v.z * v.z + v.w * v.w;
  #pragma unroll
  for (int m = 16; m > 0; m >>= 1) ss += __shfl_xor(ss, m, 32);
  if (lane == 0) {
    float n = sqrtf(ss);
    invn[row] = 1.0f / fmaxf(n, 1e-12f);
  }
}

// -------------------- K1: relW = rel_emb @ W_neighbor -----------------------
// One wave computes one 16x16 tile of C via 32 x V_WMMA_F32_16X16X4_F32.
// Out-of-range A rows: clamped pointer + multiply-by-0 mask (no divergence).
__global__ void k_rel_gemm(const float* __restrict__ rel,
                           const float* __restrict__ Wn,
                           float* __restrict__ relW, int nrows) {
  int wave = threadIdx.x >> 5;
  int lane = threadIdx.x & 31;
  int m0 = blockIdx.x * 16;
  int n0 = wave * 16;
  int lr = lane & 15;          // 0..15
  int kh = (lane >> 4) * 2;    // 0 or 2: half-wave K split (ISA 7.12.2)
  int col = n0 + lr;
  int arow = m0 + lr;
  float amask = (arow < nrows) ? 1.0f : 0.0f;
  const float* ap = rel + (size_t)(arow < nrows ? arow : 0) * D;
  v8f acc = {};
  #pragma unroll
  for (int ks = 0; ks < KSTEPS; ++ks) {
    int k = ks * 4;
    v2f a = *(const v2f*)(ap + k + kh);   // global_load_b64
    a.x *= amask;
    a.y *= amask;
    v2f b;
    b.x = Wn[(size_t)(k + kh) * D + col];
    b.y = Wn[(size_t)(k + kh + 1) * D + col];
    acc = __builtin_amdgcn_wmma_f32_16x16x4_f32(false, a, false, b,
                                                (short)0, acc, false, false);
  }
  int rbase = m0 + ((lane >> 4) ? 8 : 0);
  #pragma unroll
  for (int j = 0; j < 8; ++j) {
    int r = rbase + j;
    if (r < nrows) relW[(size_t)r * D + col] = acc[j];
  }
}

// -------- K2: hW = (ent*invn) @ W_neighbor ; agg = (ent*invn) @ W_self ------
// B operands (16 columns of each weight) live in registers; grid-stride over
// M tiles amortizes the one-time B load and gives the scheduler a long
// A-load + WMMA stream to pipeline.
__global__ void k_node_gemm(const float* __restrict__ ent,
                            const float* __restrict__ invn,
                            const float* __restrict__ Wn,
                            const float* __restrict__ Ws,
                            float* __restrict__ hW,
                            float* __restrict__ agg,
                            int ntiles) {
  int wave = threadIdx.x >> 5;
  int lane = threadIdx.x & 31;
  int n0 = wave * 16;
  int lr = lane & 15;
  int kh = (lane >> 4) * 2;
  int col = n0 + lr;

  // Preload this wave's B columns: W[:, col] for both weights (128 VGPRs).
  v2f Bn[KSTEPS], Bs[KSTEPS];
  #pragma unroll
  for (int ks = 0; ks < KSTEPS; ++ks) {
    int k = ks * 4;
    Bn[ks].x = Wn[(size_t)(k + kh) * D + col];
    Bn[ks].y = Wn[(size_t)(k + kh + 1) * D + col];
    Bs[ks].x = Ws[(size_t)(k + kh) * D + col];
    Bs[ks].y = Ws[(size_t)(k + kh + 1) * D + col];
  }

  int rsel = (lane >> 4) ? 8 : 0;
  for (int mt = blockIdx.x; mt < ntiles; mt += gridDim.x) {
    int m0 = mt * 16;
    const float* ap = ent + (size_t)(m0 + lr) * D;
    float sA = invn[m0 + lr];           // fold L2-normalize into A-tile load
    v8f accN = {};
    v8f accS = {};
    #pragma unroll
    for (int ks = 0; ks < KSTEPS; ++ks) {
      v2f a = *(const v2f*)(ap + ks * 4 + kh);   // global_load_b64
      a.x *= sA;
      a.y *= sA;
      accN = __builtin_amdgcn_wmma_f32_16x16x4_f32(false, a, false, Bn[ks],
                                                   (short)0, accN, false, false);
      accS = __builtin_amdgcn_wmma_f32_16x16x4_f32(false, a, false, Bs[ks],
                                                   (short)0, accS, false, false);
    }
    #pragma unroll
    for (int j = 0; j < 8; ++j) {
      size_t o = (size_t)(m0 + rsel + j) * D + col;
      hW[o]  = accN[j];   // neighbor-transformed features (gathered by edges)
      agg[o] = accS[j];   // seed accumulator with self-loop term h @ W_self
    }
  }
}

// -------------------- K3: edge scatter ------------------------------------
// One wave per edge; lane l handles columns l, l+32, l+64, l+96 so every
// load and every atomic instruction covers 32 consecutive dwords.
__global__ void k_edges(const float* __restrict__ hW,
                        const float* __restrict__ relW,
                        const float* __restrict__ enorm,
                        const int* __restrict__ src,
                        const int* __restrict__ dst,
                        const int* __restrict__ etype,
                        float* __restrict__ agg, int E) {
  int e = blockIdx.x * 8 + (threadIdx.x >> 5);
  int lane = threadIdx.x & 31;
  if (e >= E) return;
  int s = src[e];
  int d = dst[e];
  int t = etype[e];
  float w = enorm[e];
  const float* hs = hW + (size_t)s * D;
  const float* rs = relW + (size_t)t * D;
  float* pd = agg + (size_t)d * D;
  #pragma unroll
  for (int j = 0; j < 4; ++j) {
    int c = lane + j * 32;
    unsafeAtomicAdd(pd + c, (hs[c] + rs[c]) * w);   // global_atomic_add_f32
  }
}

// -------------------- K4: in-place rrelu (eval-mode leaky) ------------------
__global__ void k_rrelu(float* __restrict__ agg, long long n) {
  long long i = (long long)blockIdx.x * blockDim.x + threadIdx.x;
  if (i < n) {
    float x = agg[i];
    agg[i] = x >= 0.0f ? x : x * RRELU_SLOPE;
  }
}

// ---------- K5: gate = sigmoid(h_cur @ Wt + b); out = blend ----------------
__global__ void k_gate_gemm(const float* __restrict__ hc,
                            const float* __restrict__ Wt,
                            const float* __restrict__ bias,
                            const float* __restrict__ his,
                            float* __restrict__ out,
                            int ntiles) {
  int wave = threadIdx.x >> 5;
  int lane = threadIdx.x & 31;
  int n0 = wave * 16;
  int lr = lane & 15;
  int kh = (lane >> 4) * 2;
  int col = n0 + lr;

  v2f Bt[KSTEPS];
  #pragma unroll
  for (int ks = 0; ks < KSTEPS; ++ks) {
    int k = ks * 4;
    Bt[ks].x = Wt[(size_t)(k + kh) * D + col];
    Bt[ks].y = Wt[(size_t)(k + kh + 1) * D + col];
  }
  float bc = bias[col];

  int rsel = (lane >> 4) ? 8 : 0;
  for (int mt = blockIdx.x; mt < ntiles; mt += gridDim.x) {
    int m0 = mt * 16;
    const float* ap = hc + (size_t)(m0 + lr) * D;
    v8f acc = {};
    #pragma unroll
    for (int ks = 0; ks < KSTEPS; ++ks) {
      v2f a = *(const v2f*)(ap + ks * 4 + kh);   // global_load_b64
      acc = __builtin_amdgcn_wmma_f32_16x16x4_f32(false, a, false, Bt[ks],
                                                  (short)0, acc, false, false);
    }
    #pragma unroll
    for (int j = 0; j < 8; ++j) {
      size_t o = (size_t)(m0 + rsel + j) * D + col;
      float g = 1.0f / (1.0f + __expf(-(acc[j] + bc)));
      float h = hc[o];
      out[o] = g * h + (1.0f - g) * his[o];
    }
  }
}

// ---------------------------------------------------------------------------
extern "C" void kernel_launch(void* const* d_in, const int* in_sizes, int n_in,
                              void* d_out, int out_size, void* d_ws, size_t ws_size,
                              hipStream_t stream) {
  const float* ent   = (const float*)d_in[0];   // [N,128]
  const float* rel   = (const float*)d_in[1];   // [R,128]
  const float* his   = (const float*)d_in[2];   // [N,128]
  const float* Wn    = (const float*)d_in[3];   // [128,128]
  const float* Ws    = (const float*)d_in[4];   // [128,128]
  const float* Wt    = (const float*)d_in[5];   // [128,128]
  const float* bias  = (const float*)d_in[6];   // [128]
  const float* enorm = (const float*)d_in[7];   // [E]
  const int*   src   = (const int*)d_in[8];
  const int*   dst   = (const int*)d_in[9];
  const int*   etyp  = (const int*)d_in[10];
  float* out = (float*)d_out;

  const int N = in_sizes[0] / D;    // 100000
  const int R = in_sizes[1] / D;    // 1000
  const int E = in_sizes[7];        // 2000000
  const int NT = N / 16;            // 6250 M-tiles (N divisible by 16)

  // Workspace layout (floats): invn | relW | hW | agg   (~103 MB total)
  float* ws   = (float*)d_ws;
  float* invn = ws;
  float* relW = invn + (size_t)N;
  float* hW   = relW + (size_t)R * D;
  float* agg  = hW + (size_t)N * D;

  // K0: inverse row norms (8 rows per 256-thread block)
  k_invnorm<<<(N + 7) / 8, 256, 0, stream>>>(ent, invn, N);

  // K1: relW = rel_emb @ W_neighbor   (63 row tiles, 8 waves cover N=128)
  k_rel_gemm<<<(R + 15) / 16, 256, 0, stream>>>(rel, Wn, relW, R);

  // K2: hW = h @ W_neighbor, agg = h @ W_self (grid-stride over M tiles)
  int gemm_blocks = NT < 1024 ? NT : 1024;
  k_node_gemm<<<gemm_blocks, 256, 0, stream>>>(ent, invn, Wn, Ws, hW, agg, NT);

  // K3: per-edge gather + scale + atomic scatter into agg (8 edges/block)
  k_edges<<<(E + 7) / 8, 256, 0, stream>>>(hW, relW, enorm, src, dst, etyp,
                                           agg, E);

  // K4: rrelu (eval-mode leaky) in place -> h_cur
  long long nelem = (long long)N * D;
  k_rrelu<<<(unsigned)((nelem + 255) / 256), 256, 0, stream>>>(agg, nelem);

  // K5: gate GEMM + sigmoid + time-gate blend -> out
  k_gate_gemm<<<gemm_blocks, 256, 0, stream>>>(agg, Wt, bias, his, out, NT);
}